// SelfAttention_76785425318357
// MI455X (gfx1250) — compile-verified
//
#include <hip/hip_runtime.h>

// Self-attention, B=32, S=2048, D=6, fp32.
// Two-pass flash softmax; score and PV GEMMs via V_WMMA_F32_16X16X4_F32.
// Row max/sum reductions done lane-local in A-layout (1 shuffle per reduction).

typedef float v2f __attribute__((ext_vector_type(2)));
typedef float v8f __attribute__((ext_vector_type(8)));

#define FD   6
#define SEQ  2048
#define NB   32
#define NEGV (-1000000.0f)

// ---------------------------------------------------------------------------
// Kernel 1: Q/K/V projections.  Q = X @ Wq^T + bq  (torch Linear convention)
// ---------------------------------------------------------------------------
__global__ void qkv_proj_kernel(const float* __restrict__ X,
                                const float* __restrict__ Wq, const float* __restrict__ bq,
                                const float* __restrict__ Wk, const float* __restrict__ bk,
                                const float* __restrict__ Wv, const float* __restrict__ bv,
                                float* __restrict__ Q, float* __restrict__ K,
                                float* __restrict__ V, int nrows) {
    int r = blockIdx.x * blockDim.x + threadIdx.x;
    if (r >= nrows) return;
    float x[FD];
#pragma unroll
    for (int i = 0; i < FD; ++i) x[i] = X[r * FD + i];
#pragma unroll
    for (int d = 0; d < FD; ++d) {
        float q = bq[d], k = bk[d], v = bv[d];
#pragma unroll
        for (int i = 0; i < FD; ++i) {
            q = fmaf(x[i], Wq[d * FD + i], q);
            k = fmaf(x[i], Wk[d * FD + i], k);
            v = fmaf(x[i], Wv[d * FD + i], v);
        }
        Q[r * FD + d] = q;
        K[r * FD + d] = k;
        V[r * FD + d] = v;
    }
}

// ---------------------------------------------------------------------------
// Kernel 2: one wave32 per (batch, 16-row query tile).
//
// WMMA f32 16x16x4 layouts (wave32):
//   A(m,k): lane = m + 16*(k>=2), vgpr = k&1
//   B(k,n): lane = n + 16*(k>=2), vgpr = k&1
//   C(m,n): lane = n + 16*(m>=8), vgpr = m&7
//
// Pass A: online per-row max m and sum l. Scores go C-layout -> LDS ->
//   lane-per-row A-layout slices; reductions are lane-local + one shfl_xor(16).
// Pass B: recompute scores, p' = exp(s-m)/l built directly in A-layout,
//   accumulate PV with 4 WMMAs per tile. No rescale, no final division.
// ---------------------------------------------------------------------------
__global__ void __launch_bounds__(32)
attn_kernel(const float* __restrict__ Q, const float* __restrict__ K,
            const float* __restrict__ V, const int* __restrict__ amask,
            float* __restrict__ out) {
    __shared__ float pbuf[16 * 16];  // score tile bounce buffer

    const int lane = threadIdx.x;          // 0..31 (wave32)
    const int h    = lane >> 4;            // wave half
    const int n    = lane & 15;
    const int b    = blockIdx.x >> 7;      // SEQ/16 = 128 tiles per batch
    const int s0   = (blockIdx.x & 127) << 4;

    const float* Qb = Q + (size_t)b * SEQ * FD;
    const float* Kb = K + (size_t)b * SEQ * FD;
    const float* Vb = V + (size_t)b * SEQ * FD;
    const int*   mb = amask + b * SEQ;

    // Q tile in A layout: lane (h, m=n): vgpr v = Q[s0+n][2h+v] (lo), [4+2h+v] (hi)
    v2f a_lo, a_hi;
    {
        const float* qrow = Qb + (s0 + n) * FD;
        a_lo.x = qrow[2 * h];
        a_lo.y = qrow[2 * h + 1];
        a_hi.x = (h == 0) ? qrow[4] : 0.0f;
        a_hi.y = (h == 0) ? qrow[5] : 0.0f;
    }

    // Row masks for the 8 C-layout rows this lane owns (row = j + 8*h).
    bool rowm[8];
#pragma unroll
    for (int j = 0; j < 8; ++j) rowm[j] = (mb[s0 + j + 8 * h] == 0);

    // ---------------- Pass A: per-row running max/sum (row n owned by lane) --
    float m_run = -1e30f;
    float l_run = 0.0f;

    for (int t0 = 0; t0 < SEQ; t0 += 16) {
        const float* krow = Kb + (t0 + n) * FD;
        v2f b_lo, b_hi;
        b_lo.x = krow[2 * h];
        b_lo.y = krow[2 * h + 1];
        b_hi.x = (h == 0) ? krow[4] : 0.0f;
        b_hi.y = (h == 0) ? krow[5] : 0.0f;

        v8f c = {};
        c = __builtin_amdgcn_wmma_f32_16x16x4_f32(false, a_lo, false, b_lo,
                                                  (short)0, c, false, false);
        c = __builtin_amdgcn_wmma_f32_16x16x4_f32(false, a_hi, false, b_hi,
                                                  (short)0, c, false, false);

        const bool colm = (mb[t0 + n] == 0);
#pragma unroll
        for (int j = 0; j < 8; ++j)
            pbuf[(j + 8 * h) * 16 + n] = (colm || rowm[j]) ? NEGV : c[j];
        __syncthreads();

        // Lane (h,n) reads 8 elements of score row n: t_local = 4c+2h+{0,1}.
        float s8[8];
#pragma unroll
        for (int cc = 0; cc < 4; ++cc) {
            s8[2 * cc]     = pbuf[n * 16 + 4 * cc + 2 * h];
            s8[2 * cc + 1] = pbuf[n * 16 + 4 * cc + 2 * h + 1];
        }
        float mx = s8[0];
#pragma unroll
        for (int e = 1; e < 8; ++e) mx = fmaxf(mx, s8[e]);
        mx = fmaxf(mx, __shfl_xor(mx, 16, 32));     // combine with partner half
        const float mnew = fmaxf(m_run, mx);
        float sum = 0.0f;
#pragma unroll
        for (int e = 0; e < 8; ++e) sum += __expf(s8[e] - mnew);
        sum += __shfl_xor(sum, 16, 32);             // full row-16 sum
        l_run = l_run * __expf(m_run - mnew) + sum;
        m_run = mnew;
        __syncthreads();
    }
    const float inv_l = 1.0f / l_run;

    // ---------------- Pass B: recompute scores, accumulate PV ---------------
    v8f acc = {};  // output accumulator in C layout: row j+8h, col n

    for (int t0 = 0; t0 < SEQ; t0 += 16) {
        const float* krow = Kb + (t0 + n) * FD;
        v2f b_lo, b_hi;
        b_lo.x = krow[2 * h];
        b_lo.y = krow[2 * h + 1];
        b_hi.x = (h == 0) ? krow[4] : 0.0f;
        b_hi.y = (h == 0) ? krow[5] : 0.0f;

        v8f c = {};
        c = __builtin_amdgcn_wmma_f32_16x16x4_f32(false, a_lo, false, b_lo,
                                                  (short)0, c, false, false);
        c = __builtin_amdgcn_wmma_f32_16x16x4_f32(false, a_hi, false, b_hi,
                                                  (short)0, c, false, false);

        const bool colm = (mb[t0 + n] == 0);
#pragma unroll
        for (int j = 0; j < 8; ++j)
            pbuf[(j + 8 * h) * 16 + n] = (colm || rowm[j]) ? NEGV : c[j];
        __syncthreads();

        // acc += P' @ V_tile, P'[n][t] = exp(s-m)/l built in A layout directly.
#pragma unroll
        for (int cc = 0; cc < 4; ++cc) {
            v2f pa;
            pa.x = __expf(pbuf[n * 16 + 4 * cc + 2 * h] - m_run) * inv_l;
            pa.y = __expf(pbuf[n * 16 + 4 * cc + 2 * h + 1] - m_run) * inv_l;
            v2f vbv;
            if (n < FD) {
                vbv.x = Vb[(t0 + 4 * cc + 2 * h) * FD + n];
                vbv.y = Vb[(t0 + 4 * cc + 2 * h + 1) * FD + n];
            } else {
                vbv.x = 0.0f;
                vbv.y = 0.0f;
            }
            acc = __builtin_amdgcn_wmma_f32_16x16x4_f32(false, pa, false, vbv,
                                                        (short)0, acc, false, false);
        }
        __syncthreads();
    }

    // out[b, s0 + j + 8h, n] = acc[j]   (columns n < 6 are real)
    if (n < FD) {
#pragma unroll
        for (int j = 0; j < 8; ++j) {
            out[(size_t)(b * SEQ + s0 + j + 8 * h) * FD + n] = acc[j];
        }
    }
}

// ---------------------------------------------------------------------------
extern "C" void kernel_launch(void* const* d_in, const int* in_sizes, int n_in,
                              void* d_out, int out_size, void* d_ws, size_t ws_size,
                              hipStream_t stream) {
    const float* X     = (const float*)d_in[0];
    const int*   amask = (const int*)d_in[1];
    const float* Wq    = (const float*)d_in[2];
    const float* bq    = (const float*)d_in[3];
    const float* Wk    = (const float*)d_in[4];
    const float* bk    = (const float*)d_in[5];
    const float* Wv    = (const float*)d_in[6];
    const float* bv    = (const float*)d_in[7];
    float*       out   = (float*)d_out;

    const int nrows = NB * SEQ;                 // 65536
    float* Qw = (float*)d_ws;                   // 3 * nrows * 6 floats = 4.5 MB
    float* Kw = Qw + (size_t)nrows * FD;
    float* Vw = Kw + (size_t)nrows * FD;

    qkv_proj_kernel<<<(nrows + 255) / 256, 256, 0, stream>>>(
        X, Wq, bq, Wk, bk, Wv, bv, Qw, Kw, Vw, nrows);

    attn_kernel<<<NB * (SEQ / 16), 32, 0, stream>>>(Qw, Kw, Vw, amask, out);
}